// ScheduleBattery_17858474017087
// MI455X (gfx1250) — compile-verified
//
#include <hip/hip_runtime.h>

// ============================================================================
// Batched battery-scheduling QP via structure-exploiting primal-dual IPM.
//
// Exact reduction of the reference math:
//   * Q diagonal; A = +/- identity blocks  => H = Q + A^T diag(d) A is DIAGONAL
//   * Ae is the battery-dynamics operator (<=4 nnz/row, chain overlap)
//       => Schur complement S = Ae H^-1 Ae^T is TRIDIAGONAL 48x48 SPD
//   * 192x192 KKT solve == Thomas algorithm on S (O(T) per batch per iter)
//
// Mapping: 1 workgroup (64 threads, 2x wave32) per batch problem, all 30 IPM
// iterations fused in one kernel. CDNA5 specifics used:
//   * GLOBAL_LOAD_ASYNC_TO_LDS_B32 + s_wait_asynccnt (ASYNCcnt path) for the
//     global->LDS price copy (no VGPR round-trip)
//   * wave32 __shfl_xor butterfly reductions (ds-swizzle path), 1 barrier each
//   * v_rcp_f32 + 1 Newton step on the serial Thomas critical path
// WMMA is intentionally absent: diagonal/tridiagonal algebra has no dense-tile
// mapping; the dense WMMA formulation would do ~5000x more flops.
// ============================================================================

#define TT    48
#define NVAR  144   // 3T
#define MCON  288   // 6T
#define NBATCH 1024
#define ITERS 30
#define EPSQ  0.01f
#define EFF   0.95f
#define LAMW  0.1f
#define SIGMA 0.1f
#define BIGV  1e10f

__device__ __forceinline__ float fast_rcp(float x) {
    // v_rcp_f32 (~1 ulp) + one Newton-Raphson step -> ~correctly rounded,
    // much shorter latency than the compiler's full division sequence.
    float r = __builtin_amdgcn_rcpf(x);
    r = __builtin_fmaf(__builtin_fmaf(-x, r, 1.0f), r, r);
    return r;
}

// Block reduction: wave32 butterfly via __shfl_xor, then one 2-slot LDS
// exchange across the two waves. Exactly one __syncthreads.
__device__ __forceinline__ float block_sum(float v, float* red2, int tid) {
    #pragma unroll
    for (int off = 16; off > 0; off >>= 1) v += __shfl_xor(v, off, 32);
    if ((tid & 31) == 0) red2[tid >> 5] = v;
    __syncthreads();
    return red2[0] + red2[1];
}
__device__ __forceinline__ float block_min(float v, float* red2, int tid) {
    #pragma unroll
    for (int off = 16; off > 0; off >>= 1) v = fminf(v, __shfl_xor(v, off, 32));
    if ((tid & 31) == 0) red2[tid >> 5] = v;
    __syncthreads();
    return fminf(red2[0], red2[1]);
}

__global__ __launch_bounds__(64)
void schedule_battery_ipm(const float* __restrict__ log_prices,
                          float* __restrict__ out)
{
    const int b   = blockIdx.x;
    const int tid = threadIdx.x;

    // Persistent IPM state (per problem) + per-iteration temporaries in LDS.
    __shared__ float z[NVAR], c[NVAR];
    __shared__ float s[MCON], lam[MCON], nu[TT];
    __shared__ float d[MCON], rp[MCON], rcomp[MCON], w[MCON];
    __shared__ float dsv[MCON], dlamv[MCON];
    __shared__ float hinv[NVAR], rhsz[NVAR], g[NVAR], dzv[NVAR];
    __shared__ float Sdiag[TT], Ssub[TT], Srhs[TT], dnuv[TT];
    __shared__ float lpS[TT];
    __shared__ float red2[2];

    const float* lpb = log_prices + b * TT;

    // ---- async global->LDS copy of this problem's 48 log-prices ------------
    // GVS mode: mem_addr = SGPR64 base + VGPR32 offset; VDST VGPR = LDS addr.
    if (tid < TT) {
        unsigned lds_off = (unsigned)(unsigned long long)(&lpS[tid]);
        unsigned goff    = (unsigned)(tid * 4);
        asm volatile("global_load_async_to_lds_b32 %0, %1, %2"
                     :: "v"(lds_off), "v"(goff),
                        "s"((unsigned long long)lpb)
                     : "memory");
    }
    asm volatile("s_wait_asynccnt 0x0" ::: "memory");
    __syncthreads();

    // ---- init: c = [exp(lp), -exp(lp), -LAM*BCAP], z=0, s=1, lam=1, nu=0 ----
    if (tid < TT) {
        float p = expf(lpS[tid]);
        c[tid]        =  p;
        c[TT + tid]   = -p;
        c[2*TT + tid] = -LAMW * 1.0f;   // c_tail = -LAM*BCAP
        nu[tid] = 0.0f;
    }
    for (int v = tid; v < NVAR; v += 64) z[v] = 0.0f;
    for (int j = tid; j < MCON; j += 64) { s[j] = 1.0f; lam[j] = 1.0f; }
    __syncthreads();

    for (int it = 0; it < ITERS; ++it) {
        // ---------- pass 1 over constraints: d, rp, and partial s.lam ----------
        float acc = 0.0f;
        for (int j = tid; j < MCON; j += 64) {
            int blk    = j / 96;
            int within = j - blk * 96;
            int isLo   = within >= TT;
            int k      = isLo ? within - TT : within;
            int v      = blk * TT + k;
            float sj = s[j], lj = lam[j];
            float bj  = isLo ? 0.0f : (blk == 0 ? 0.5f : (blk == 1 ? 0.2f : 1.0f));
            float sgn = isLo ? -1.0f : 1.0f;
            acc  += sj * lj;
            d[j]  = lj / sj;
            rp[j] = sgn * z[v] + sj - bj;            // (Az)_j + s - b
        }
        const float mu    = block_sum(acc, red2, tid) * (1.0f / (float)MCON);
        const float sigmu = SIGMA * mu;

        // ---------- pass 2 over constraints: rcomp, w = (-rcomp+lam*rp)/s ----
        for (int j = tid; j < MCON; j += 64) {
            float sj = s[j], lj = lam[j];
            float rcj = sj * lj - sigmu;
            rcomp[j] = rcj;
            w[j]     = (-rcj + lj * rp[j]) / sj;
        }
        __syncthreads();

        // ---------- per-variable: rz, H diag, rhs_z, g = H^-1 rhs_z ----------
        for (int v = tid; v < NVAR; v += 64) {
            int blkv = v / TT;
            int kk   = v - blkv * TT;
            int up   = v + blkv * TT;     // upper-bound row of var v
            int lo   = up + TT;           // lower-bound row of var v
            // (Ae^T nu)_v from sparse battery-dynamics rows
            float aet;
            if (blkv == 0)      aet = (kk < TT-1) ?  EFF * nu[kk+1] : 0.0f;
            else if (blkv == 1) aet = (kk < TT-1) ? -nu[kk+1]       : 0.0f;
            else                aet = (kk < TT-1 ? nu[kk+1] : 0.0f)
                                    + (kk == 0 ? nu[0] : -nu[kk]);
            float qv = (v < 2*TT) ? EPSQ : LAMW;
            float rz = qv * z[v] + c[v] + (lam[up] - lam[lo]) + aet;
            float Hv = qv + d[up] + d[lo];           // H is diagonal
            float hi = 1.0f / Hv;
            float rv = -rz - (w[up] - w[lo]);        // rhs_z
            hinv[v] = hi; rhsz[v] = rv; g[v] = hi * rv;
        }
        __syncthreads();

        // ---------- tridiagonal Schur system S dnu = Ae*g + re ----------
        if (tid < TT) {
            int r = tid;
            if (r == 0) {
                Sdiag[0] = hinv[2*TT];
                Ssub[0]  = 0.0f;
                Srhs[0]  = g[2*TT] + (z[2*TT] - 0.5f);      // re0 = z3[0]-BCAP/2
            } else {
                float re_r = EFF * z[r-1] - z[TT-1+r] + z[2*TT-1+r] - z[2*TT+r];
                Srhs[r]  = EFF * g[r-1] - g[TT-1+r] + g[2*TT-1+r] - g[2*TT+r] + re_r;
                Sdiag[r] = (EFF*EFF) * hinv[r-1] + hinv[TT-1+r]
                         + hinv[2*TT-1+r] + hinv[2*TT+r];
                Ssub[r]  = (r == 1) ? hinv[2*TT] : -hinv[2*TT-1+r];
            }
        }
        __syncthreads();

        // ---------- Thomas solve (serial, lane 0; rcp-based pivots) ----------
        if (tid == 0) {
            float invprev = fast_rcp(Sdiag[0]);
            float rprev   = Srhs[0];
            Sdiag[0] = invprev;                 // cache 1/diag for back-sweep
            for (int r = 1; r < TT; ++r) {
                float sb = Ssub[r];
                float wf = sb * invprev;
                float dr = Sdiag[r] - wf * sb;
                float rr = Srhs[r]  - wf * rprev;
                float iv = fast_rcp(dr);
                Sdiag[r] = iv; Srhs[r] = rr;
                invprev = iv; rprev = rr;
            }
            float x = Srhs[TT-1] * Sdiag[TT-1];
            dnuv[TT-1] = x;
            for (int r = TT-2; r >= 0; --r) {
                x = (Srhs[r] - Ssub[r+1] * x) * Sdiag[r];
                dnuv[r] = x;
            }
        }
        __syncthreads();

        // ---------- dz = H^-1 (rhs_z - Ae^T dnu) ----------
        for (int v = tid; v < NVAR; v += 64) {
            int blkv = v / TT;
            int kk   = v - blkv * TT;
            float aet;
            if (blkv == 0)      aet = (kk < TT-1) ?  EFF * dnuv[kk+1] : 0.0f;
            else if (blkv == 1) aet = (kk < TT-1) ? -dnuv[kk+1]       : 0.0f;
            else                aet = (kk < TT-1 ? dnuv[kk+1] : 0.0f)
                                    + (kk == 0 ? dnuv[0] : -dnuv[kk]);
            dzv[v] = hinv[v] * (rhsz[v] - aet);
        }
        __syncthreads();

        // ---------- ds, dlam, fraction-to-boundary ratio ----------
        float rmin = BIGV;
        for (int j = tid; j < MCON; j += 64) {
            int blk    = j / 96;
            int within = j - blk * 96;
            int isLo   = within >= TT;
            int k      = isLo ? within - TT : within;
            int v      = blk * TT + k;
            float sgn  = isLo ? -1.0f : 1.0f;
            float dsj  = -rp[j] - sgn * dzv[v];         // -rp - (A dz)_j
            float sj = s[j], lj = lam[j];
            float dlj  = (-rcomp[j] - lj * dsj) / sj;
            dsv[j] = dsj; dlamv[j] = dlj;
            if (dsj < 0.0f) rmin = fminf(rmin, -sj / dsj);
            if (dlj < 0.0f) rmin = fminf(rmin, -lj / dlj);
        }
        const float alpha = fminf(1.0f, 0.99f * block_min(rmin, red2, tid));

        // ---------- update state ----------
        for (int v = tid; v < NVAR; v += 64) z[v] += alpha * dzv[v];
        for (int j = tid; j < MCON; j += 64) {
            s[j]   += alpha * dsv[j];
            lam[j] += alpha * dlamv[j];
        }
        if (tid < TT) nu[tid] += alpha * dnuv[tid];
        __syncthreads();
    }

    // ---------- output z (1024 x 144) ----------
    float* ob = out + b * NVAR;
    for (int v = tid; v < NVAR; v += 64) ob[v] = z[v];
}

extern "C" void kernel_launch(void* const* d_in, const int* in_sizes, int n_in,
                              void* d_out, int out_size, void* d_ws, size_t ws_size,
                              hipStream_t stream) {
    (void)in_sizes; (void)n_in; (void)d_ws; (void)ws_size; (void)out_size;
    const float* log_prices = (const float*)d_in[0];  // [1024, 48] f32
    float* out = (float*)d_out;                       // [1024, 144] f32
    schedule_battery_ipm<<<NBATCH, 64, 0, stream>>>(log_prices, out);
}